// Dep_Context_60464549593215
// MI455X (gfx1250) — compile-verified
//
#include <hip/hip_runtime.h>
#include <hip/hip_bf16.h>

typedef float v2f __attribute__((ext_vector_type(2)));
typedef float v8f __attribute__((ext_vector_type(8)));

#define NB   8
#define CCH  256
#define HID  10
#define HW   3600
#define KP   20      // K = 18 padded to 20 (5 WMMA k-steps of 4)
#define NTIL 225     // 3600 / 16

__device__ __forceinline__ void coord8(int sp, float* f) {
    int row = sp / 60, col = sp - row * 60;
    const float s = 2.0f / 60.0f;
    float xmin = col * s - 1.0f;
    float xmax = (col + 1) * s - 1.0f;
    float ymin = row * s - 1.0f;
    float ymax = (row + 1) * s - 1.0f;
    f[0] = xmin; f[1] = ymin; f[2] = xmax; f[3] = ymax;
    f[4] = 0.5f * (xmin + xmax); f[5] = 0.5f * (ymin + ymax);
    f[6] = 1.0f / 60.0f; f[7] = 1.0f / 60.0f;
}

// LDS byte offset of a pointer into a __shared__ array (generic -> addrspace(3) -> int)
__device__ __forceinline__ unsigned ldsoff(const float* p) {
    return (unsigned)(unsigned long long)(__attribute__((address_space(3))) const float*)p;
}

// CDNA5 async global->LDS copy, 16 bytes per lane, tracked with ASYNCcnt
__device__ __forceinline__ void async_copy_b128(unsigned lds_byte_addr, const float* gaddr) {
    asm volatile("global_load_async_to_lds_b128 %0, %1, off"
                 :: "v"(lds_byte_addr), "v"(gaddr) : "memory");
}
__device__ __forceinline__ void wait_async0() {
    asm volatile("s_wait_asynccnt 0x0" ::: "memory");
}

// ---- K1: xff_sum = sum_d hu^2, plus per-batch spatial max ----
__global__ __launch_bounds__(256) void athu_kernel(
    const float* __restrict__ hu, float* __restrict__ xff, float* __restrict__ humax) {
    int b = blockIdx.x;
    const float* h = hu + (size_t)b * HID * HW;
    float lmax = -3.402823466e38f;
    for (int t = threadIdx.x; t < HW; t += 256) {
        float s = 0.0f;
#pragma unroll
        for (int d = 0; d < HID; ++d) { float v = h[(size_t)d * HW + t]; s = fmaf(v, v, s); }
        xff[(size_t)b * HW + t] = s;
        lmax = fmaxf(lmax, s);
    }
    __shared__ float red[256];
    red[threadIdx.x] = lmax;
    __syncthreads();
    for (int off = 128; off > 0; off >>= 1) {
        if (threadIdx.x < off) red[threadIdx.x] = fmaxf(red[threadIdx.x], red[threadIdx.x + off]);
        __syncthreads();
    }
    if (threadIdx.x == 0) humax[b] = red[0];
}

// ---- K2: project1 = [p_fea^T | coord] @ W  -> (n, HW, KP) with zero pads ----
__global__ __launch_bounds__(256) void project1_kernel(
    const float* __restrict__ p_fea, const float* __restrict__ W, float* __restrict__ P1) {
    __shared__ float sW[(CCH + 8) * 18];
    for (int i = threadIdx.x; i < (CCH + 8) * 18; i += 256) sW[i] = W[i];
    __syncthreads();
    int b = blockIdx.y;
    int sp = blockIdx.x * 256 + threadIdx.x;
    if (sp >= HW) return;
    const float* pf = p_fea + (size_t)b * CCH * HW + sp;
    float acc[18];
#pragma unroll
    for (int d = 0; d < 18; ++d) acc[d] = 0.0f;
    for (int k = 0; k < CCH; ++k) {
        float x = pf[(size_t)k * HW];
        const float* wr = &sW[k * 18];
#pragma unroll
        for (int d = 0; d < 18; ++d) acc[d] = fmaf(x, wr[d], acc[d]);
    }
    float cf[8]; coord8(sp, cf);
#pragma unroll
    for (int j = 0; j < 8; ++j) {
        const float* wr = &sW[(CCH + j) * 18];
#pragma unroll
        for (int d = 0; d < 18; ++d) acc[d] = fmaf(cf[j], wr[d], acc[d]);
    }
    float* out = P1 + ((size_t)b * HW + sp) * KP;
#pragma unroll
    for (int d = 0; d < 18; ++d) out[d] = acc[d];
    out[18] = 0.0f; out[19] = 0.0f;
}

// ---- K3: rightT[t][d] = [att_hu*hu | coord], (n, HW, KP) with zero pads ----
__global__ __launch_bounds__(256) void right_kernel(
    const float* __restrict__ hu, const float* __restrict__ xff,
    const float* __restrict__ humax, float* __restrict__ RT) {
    int b = blockIdx.y;
    int t = blockIdx.x * 256 + threadIdx.x;
    if (t >= HW) return;
    float att = xff[(size_t)b * HW + t] / humax[b];
    const float* h = hu + (size_t)b * HID * HW + t;
    float* out = RT + ((size_t)b * HW + t) * KP;
#pragma unroll
    for (int d = 0; d < HID; ++d) out[d] = att * h[(size_t)d * HW];
    float cf[8]; coord8(t, cf);
#pragma unroll
    for (int j = 0; j < 8; ++j) out[HID + j] = cf[j];
    out[18] = 0.0f; out[19] = 0.0f;
}

// ---- K4: project2 row-max. 8 waves/block share each B tile through LDS       ----
// ---- (async global->LDS, double buffered); 5x v_wmma_f32_16x16x4_f32 per tile ----
__global__ __launch_bounds__(256) void gemm_rowmax_kernel(
    const float* __restrict__ P1, const float* __restrict__ RT, float* __restrict__ attc) {
    __shared__ __align__(16) float sB[2][16 * KP];   // two 1.28 KB B tiles

    const int tid  = threadIdx.x;
    const int wave = tid >> 5;
    const int lane = tid & 31;
    const int b    = blockIdx.y;
    const int rowtile = blockIdx.x * 8 + wave;
    const int rt = (rowtile < NTIL) ? rowtile : (NTIL - 1); // clamp: all waves stay in barriers

    const float* A  = P1 + (size_t)b * HW * KP;
    const float* Bm = RT + (size_t)b * HW * KP;

    const int mrow  = rt * 16 + (lane & 15);
    const int khalf = (lane >> 4) * 2;     // lanes 0-15: K=0,1 ; lanes 16-31: K=2,3

    v2f a[5];
#pragma unroll
    for (int f = 0; f < 5; ++f)
        a[f] = *(const v2f*)(A + (size_t)mrow * KP + f * 4 + khalf);

    // loader role: 80 threads fetch 16 rows x 20 floats as 5 x b128 per row
    const bool loader = (tid < 80);
    const int  lrow = tid / 5;             // 0..15
    const int  lq   = tid - lrow * 5;      // 0..4
    const float* gsrc = Bm + lrow * KP + lq * 4;
    unsigned ldst0 = ldsoff(&sB[0][lrow * KP + lq * 4]);
    unsigned ldst1 = ldsoff(&sB[1][lrow * KP + lq * 4]);

    if (loader) async_copy_b128(ldst0, gsrc);    // prefetch tile 0

    v8f mx;
#pragma unroll
    for (int r = 0; r < 8; ++r) mx[r] = -3.402823466e38f;

    for (int tc = 0; tc < NTIL; ++tc) {
        wait_async0();          // issuing waves: tile (tc) landed in LDS
        __syncthreads();        // publish tile (tc); prior reads of other buffer done
        if (loader && (tc + 1 < NTIL))
            async_copy_b128((tc & 1) ? ldst0 : ldst1,
                            gsrc + (size_t)(tc + 1) * 16 * KP);
        const float* bbase = &sB[tc & 1][(lane & 15) * KP + khalf];
        v8f c = {};
#pragma unroll
        for (int f = 0; f < 5; ++f) {
            v2f bv = *(const v2f*)(bbase + f * 4);
            c = __builtin_amdgcn_wmma_f32_16x16x4_f32(false, a[f], false, bv,
                                                      (short)0, c, false, false);
        }
#pragma unroll
        for (int r = 0; r < 8; ++r) mx[r] = fmaxf(mx[r], c[r]);
    }

    // reduce over N (16 lanes within each half); xor 1,2,4,8 stays inside a half
#pragma unroll
    for (int off = 1; off < 16; off <<= 1) {
#pragma unroll
        for (int r = 0; r < 8; ++r)
            mx[r] = fmaxf(mx[r], __shfl_xor(mx[r], off, 32));
    }
    if (rowtile < NTIL && (lane & 15) == 0) {
        int mbase = rowtile * 16 + (lane >> 4) * 8;  // lane0 -> rows 0-7, lane16 -> rows 8-15
#pragma unroll
        for (int r = 0; r < 8; ++r) attc[(size_t)b * HW + mbase + r] = mx[r];
    }
}

// ---- K5: per-batch max of att_context ----
__global__ __launch_bounds__(256) void amax_kernel(
    const float* __restrict__ attc, float* __restrict__ amax) {
    int b = blockIdx.x;
    float lmax = -3.402823466e38f;
    for (int t = threadIdx.x; t < HW; t += 256)
        lmax = fmaxf(lmax, attc[(size_t)b * HW + t]);
    __shared__ float red[256];
    red[threadIdx.x] = lmax;
    __syncthreads();
    for (int off = 128; off > 0; off >>= 1) {
        if (threadIdx.x < off) red[threadIdx.x] = fmaxf(red[threadIdx.x], red[threadIdx.x + off]);
        __syncthreads();
    }
    if (threadIdx.x == 0) amax[b] = red[0];
}

// ---- K6: out = (att_ctx/amax) * p_fea * (1 - xff/humax) ----
__global__ __launch_bounds__(256) void final_kernel(
    const float* __restrict__ p_fea, const float* __restrict__ xff,
    const float* __restrict__ humax, const float* __restrict__ attc,
    const float* __restrict__ amax, float* __restrict__ out) {
    size_t idx = (size_t)blockIdx.x * 256 + threadIdx.x;
    int b  = (int)(idx / ((size_t)CCH * HW));
    int sp = (int)(idx % HW);
    float att_hu = xff[(size_t)b * HW + sp] / humax[b];
    float dep    = attc[(size_t)b * HW + sp] / amax[b];
    out[idx] = dep * p_fea[idx] * (1.0f - att_hu);
}

extern "C" void kernel_launch(void* const* d_in, const int* in_sizes, int n_in,
                              void* d_out, int out_size, void* d_ws, size_t ws_size,
                              hipStream_t stream) {
    const float* p_fea = (const float*)d_in[0];   // (8,256,60,60)
    const float* hu    = (const float*)d_in[1];   // (8,10,60,60)
    const float* W     = (const float*)d_in[2];   // (264,18)
    float* out = (float*)d_out;

    float* xff   = (float*)d_ws;                      // NB*HW
    float* humax = xff + (size_t)NB * HW;             // NB
    float* P1    = humax + 8;                         // NB*HW*KP (offset 16B-aligned)
    float* RT    = P1 + (size_t)NB * HW * KP;         // NB*HW*KP (offset 16B-aligned)
    float* attc  = RT + (size_t)NB * HW * KP;         // NB*HW
    float* amax  = attc + (size_t)NB * HW;            // NB
    (void)in_sizes; (void)n_in; (void)out_size; (void)ws_size;

    athu_kernel       <<<dim3(NB),       256, 0, stream>>>(hu, xff, humax);
    project1_kernel   <<<dim3(15, NB),   256, 0, stream>>>(p_fea, W, P1);
    right_kernel      <<<dim3(15, NB),   256, 0, stream>>>(hu, xff, humax, RT);
    gemm_rowmax_kernel<<<dim3(29, NB),   256, 0, stream>>>(P1, RT, attc);  // 29*8 waves >= 225 tiles
    amax_kernel       <<<dim3(NB),       256, 0, stream>>>(attc, amax);
    final_kernel      <<<dim3(28800),    256, 0, stream>>>(p_fea, xff, humax, attc, amax, out);
}